// EIB_GNN1D_L_Nested_Power_86878598463993
// MI455X (gfx1250) — compile-verified
//
#include <hip/hip_runtime.h>

typedef float v2f __attribute__((ext_vector_type(2)));
typedef float v8f __attribute__((ext_vector_type(8)));

#define LCNT 64
#define BCNT 256

// ---------------- pack: Hhat -> X0[n][b][c]  (c = i*32 + k*4 + u) ----------------
__global__ void pack_kernel(const float* __restrict__ Hre,
                            const float* __restrict__ Him,
                            float* __restrict__ X0) {
  unsigned t = blockIdx.x * 256u + threadIdx.x;   // 1,048,576 total
  unsigned c = t & 63u;
  unsigned b = (t >> 6) & 255u;
  unsigned n = t >> 14;
  const float* __restrict__ H = (c < 32u) ? Hre : Him;
  X0[t] = H[(size_t)b * 2048u + (size_t)n * 32u + (c & 31u)];
}

// ---------------- WMMA f32 GEMM: Y[n][b][o] = sum_d P[n][o][d]*X[n][b][d] ----------------
// WG = 256 thr = 8 waves. WG tile: M=128, N=64. Wave tile: 64(M) x 16(N), 4 accumulators.
// Double-buffered LDS staging, register-prefetched globals, 1 barrier per K-block.
__global__ __launch_bounds__(256) void gemm_kernel(
    const float* __restrict__ P, const float* __restrict__ X,
    float* __restrict__ Y, int KDIM, int M_P, int M_OUT, int relu)
{
  const int n  = blockIdx.z;
  const int m0 = blockIdx.y * 128;
  const int b0 = blockIdx.x * 64;
  const int t  = threadIdx.x;
  const int wave = t >> 5;
  const int lane = t & 31;
  const int wn = wave & 3;      // N subtile (16 wide)
  const int wm = wave >> 2;     // M half (64 tall)
  const int hi = lane >> 4;
  const int lr = lane & 15;

  // LDS: P double-buffer 2x[128][17]=4352 f, XT double-buffer 2x[8][66] float2 = 2112 f
  // epilogue reuses as [64][129] = 8256 f
  __shared__ float smem[8448];
  float2* xtBase = (float2*)(smem + 4352);

  const float* __restrict__ Pn = P + (size_t)n * M_P * KDIM;
  const float* __restrict__ Xn = X + (size_t)n * BCNT * KDIM;

  v8f acc[4];
  #pragma unroll
  for (int i = 0; i < 4; ++i) acc[i] = (v8f){0.f,0.f,0.f,0.f,0.f,0.f,0.f,0.f};

  // staging thread mapping (all rows in-bounds by construction; no exec guards)
  const int prow = t >> 1;             // 0..127
  const int pk   = (t & 1) * 8;        // 0 or 8
  const int xrow = t >> 2;             // 0..63
  const int xk   = (t & 3) * 4;        // 0,4,8,12

  const float* __restrict__ pSrc = Pn + (size_t)(m0 + prow) * KDIM + pk;
  const float* __restrict__ xSrc = Xn + (size_t)(b0 + xrow) * KDIM + xk;

  const int nIter = KDIM >> 4;

  // prologue: stage K-block 0 into buffer 0
  {
    float4 pv0 = *(const float4*)(pSrc);
    float4 pv1 = *(const float4*)(pSrc + 4);
    float4 xv  = *(const float4*)(xSrc);
    float* dp = smem + prow * 17 + pk;
    dp[0]=pv0.x; dp[1]=pv0.y; dp[2]=pv0.z; dp[3]=pv0.w;
    dp[4]=pv1.x; dp[5]=pv1.y; dp[6]=pv1.z; dp[7]=pv1.w;
    xtBase[(xk >> 1) * 66 + xrow]       = make_float2(xv.x, xv.y);
    xtBase[((xk >> 1) + 1) * 66 + xrow] = make_float2(xv.z, xv.w);
  }
  __syncthreads();

  for (int it = 0; it < nIter; ++it) {
    const int cur = it & 1;
    const bool more = (it + 1) < nIter;

    // prefetch next K-block into registers (overlaps with WMMA below)
    float4 npv0 = make_float4(0.f,0.f,0.f,0.f);
    float4 npv1 = npv0, nxv = npv0;
    if (more) {
      const float* ps = pSrc + (size_t)(it + 1) * 16;
      npv0 = *(const float4*)(ps);
      npv1 = *(const float4*)(ps + 4);
      nxv  = *(const float4*)(xSrc + (size_t)(it + 1) * 16);
    }

    const float*  bP = smem + cur * 2176;
    const float2* bX = xtBase + cur * 528;
    #pragma unroll
    for (int kk = 0; kk < 16; kk += 4) {
      float2 bfv = bX[((kk >> 1) + hi) * 66 + wn * 16 + lr];  // single ds_load_b64
      v2f bf; bf.x = bfv.x; bf.y = bfv.y;
      #pragma unroll
      for (int tM = 0; tM < 4; ++tM) {
        const float* pa = bP + (wm * 64 + tM * 16 + lr) * 17 + kk + 2 * hi;
        v2f a; a.x = pa[0]; a.y = pa[1];
        acc[tM] = __builtin_amdgcn_wmma_f32_16x16x4_f32(false, a, false, bf,
                                                        (short)0, acc[tM], false, false);
      }
    }

    if (more) {
      float*  wP = smem + (cur ^ 1) * 2176;
      float2* wX = xtBase + (cur ^ 1) * 528;
      float* dp = wP + prow * 17 + pk;
      dp[0]=npv0.x; dp[1]=npv0.y; dp[2]=npv0.z; dp[3]=npv0.w;
      dp[4]=npv1.x; dp[5]=npv1.y; dp[6]=npv1.z; dp[7]=npv1.w;
      wX[(xk >> 1) * 66 + xrow]       = make_float2(nxv.x, nxv.y);
      wX[((xk >> 1) + 1) * 66 + xrow] = make_float2(nxv.z, nxv.w);
    }
    __syncthreads();
  }

  // epilogue through LDS for coalesced stores; fused ReLU
  float* ldsO = smem;  // [64][129]
  #pragma unroll
  for (int tM = 0; tM < 4; ++tM) {
    #pragma unroll
    for (int r = 0; r < 8; ++r) {
      float v = acc[tM][r];
      if (relu) v = fmaxf(v, 0.f);
      ldsO[(wn * 16 + lr) * 129 + wm * 64 + tM * 16 + r + 8 * hi] = v;
    }
  }
  __syncthreads();
  {
    const int row = t >> 2;          // 0..63 (b_local)
    const int c0  = (t & 3) * 32;    // 0..96
    float* __restrict__ dst = Y + (size_t)n * BCNT * M_OUT
                                + (size_t)(b0 + row) * M_OUT + m0 + c0;
    const float* src = ldsO + row * 129 + c0;
    #pragma unroll
    for (int i = 0; i < 32; ++i)
      if (m0 + c0 + i < M_OUT) dst[i] = src[i];
  }
}

// ---------------- BN partial stats (deterministic, per-n partials) ----------------
__global__ __launch_bounds__(256) void bn_stats_kernel(
    const float* __restrict__ Y, float* __restrict__ psum, float* __restrict__ psq)
{
  const int n = blockIdx.x;       // 64
  const int t = threadIdx.x;      // 256 threads x 4 channels
  const float* __restrict__ base = Y + (size_t)n * BCNT * 1024 + t * 4;
  float s0=0.f,s1=0.f,s2=0.f,s3=0.f,q0=0.f,q1=0.f,q2=0.f,q3=0.f;
  for (int b = 0; b < BCNT; ++b) {
    float4 v = *(const float4*)(base + (size_t)b * 1024);
    s0 += v.x; q0 += v.x*v.x;
    s1 += v.y; q1 += v.y*v.y;
    s2 += v.z; q2 += v.z*v.z;
    s3 += v.w; q3 += v.w*v.w;
  }
  *(float4*)(psum + (size_t)n*1024 + t*4) = make_float4(s0,s1,s2,s3);
  *(float4*)(psq  + (size_t)n*1024 + t*4) = make_float4(q0,q1,q2,q3);
}

__global__ void bn_finalize_kernel(const float* __restrict__ psum,
                                   const float* __restrict__ psq,
                                   const float* __restrict__ w,
                                   const float* __restrict__ bias,
                                   float* __restrict__ scale,
                                   float* __restrict__ shift)
{
  const int j = blockIdx.x * 256 + threadIdx.x;  // 1024 channels
  float s = 0.f, q = 0.f;
  for (int n = 0; n < LCNT; ++n) { s += psum[n*1024 + j]; q += psq[n*1024 + j]; }
  const float inv = 1.0f / 16384.0f;
  float mu   = s * inv;
  float var  = q * inv - mu * mu;
  float rstd = 1.0f / sqrtf(var + 1e-5f);
  float sc   = w[j] * rstd;
  scale[j] = sc;
  shift[j] = bias[j] - mu * sc;
}

// ---------------- fused BN-apply + noise MI + write next-layer input ----------------
__global__ __launch_bounds__(256) void noise_kernel(
    const float* __restrict__ Y,       // (L,256,1024) post-relu, pre-BN
    const float* __restrict__ scale,
    const float* __restrict__ shift,
    const float* __restrict__ noise,   // (4, 16384, 512)  idx = s, b*L+n, j
    float* __restrict__ Xnext,         // (L,256,512)  normalized mean half
    float* __restrict__ Ipart)         // (256, 64)
{
  const int n = blockIdx.x;   // 64
  const int b = blockIdx.y;   // 256
  const int t = threadIdx.x;  // 256
  const float* __restrict__ y  = Y     + ((size_t)n * BCNT + b) * 1024;
  float* __restrict__       xn = Xnext + ((size_t)n * BCNT + b) * 512;
  const size_t nbase = ((size_t)b * LCNT + n) * 512;
  float acc = 0.f;
  #pragma unroll
  for (int u = 0; u < 2; ++u) {
    const int jj = t + u * 256;
    float m  = y[jj]       * scale[jj]       + shift[jj];
    float sr = y[jj + 512] * scale[jj + 512] + shift[jj + 512];
    float sp = fmaxf(sr, 0.f) + log1pf(expf(-fabsf(sr)));   // softplus
    float sd = sp + 1e-10f;
    xn[jj] = m;
    float a = 0.f;
    #pragma unroll
    for (int s = 0; s < 4; ++s) {
      float nv = noise[(size_t)s * (16384u * 512u) + nbase + jj];
      float z = m + sd * nv;
      a += 0.5f * z * z - 0.5f * nv * nv;
    }
    acc += 0.25f * a - logf(sd);
  }
  __shared__ float red[256];
  red[t] = acc;
  __syncthreads();
  for (int o = 128; o > 0; o >>= 1) {
    if (t < o) red[t] += red[t + o];
    __syncthreads();
  }
  if (t == 0) Ipart[(size_t)b * LCNT + n] = red[0];
}

// ---------------- final normalization: Fre / Fim ----------------
__global__ void final_kernel(const float* __restrict__ Y2,  // (L,256,72)
                             float* __restrict__ out)       // Fre@0, Fim@65536
{
  const int b = blockIdx.x;   // 256
  const int l = threadIdx.x;  // 64
  __shared__ float shp[64][8];
  __shared__ float pns[8];
  const float* __restrict__ y = Y2 + ((size_t)l * BCNT + b) * 72;
  float pu[8];
  #pragma unroll
  for (int k = 0; k < 8; ++k) { float v = y[64 + k]; pu[k] = v; shp[l][k] = v * v; }
  __syncthreads();
  if (l < 8) {
    float s = 0.f;
    for (int i = 0; i < 64; ++i) s += shp[i][l];
    pns[l] = sqrtf(s);                 // norm over L per (b,k)
  }
  __syncthreads();
  float f1[32], f2[32], ss[8];
  #pragma unroll
  for (int k = 0; k < 8; ++k) ss[k] = 0.f;
  #pragma unroll
  for (int o = 0; o < 32; ++o) {
    f1[o] = y[o]; f2[o] = y[32 + o];
    ss[o >> 2] += f1[o] * f1[o] + f2[o] * f2[o];
  }
  float mk[8];
  #pragma unroll
  for (int k = 0; k < 8; ++k)
    mk[k] = (1.0f / sqrtf(ss[k])) * (pu[k] / pns[k] * 8.0f);   // 8 = sqrt(L)
  const size_t base = (size_t)b * 2048 + (size_t)l * 32;
  #pragma unroll
  for (int o = 0; o < 32; ++o) {
    out[base + o]         = f1[o] * mk[o >> 2];
    out[65536 + base + o] = f2[o] * mk[o >> 2];
  }
}

__global__ void iout_kernel(const float* __restrict__ Ip0,
                            const float* __restrict__ Ip1,
                            float* __restrict__ outI)
{
  const int b = threadIdx.x;  // 256
  float s = 0.f;
  for (int n = 0; n < LCNT; ++n) s += Ip0[b * LCNT + n] + Ip1[b * LCNT + n];
  outI[b] = s;
}

extern "C" void kernel_launch(void* const* d_in, const int* in_sizes, int n_in,
                              void* d_out, int out_size, void* d_ws, size_t ws_size,
                              hipStream_t stream) {
  (void)in_sizes; (void)n_in; (void)out_size; (void)ws_size;
  const float* Hre = (const float*)d_in[0];
  const float* Him = (const float*)d_in[1];
  const float* P0  = (const float*)d_in[2];   // (64,1024,64)
  const float* P1  = (const float*)d_in[3];   // (64,1024,512)
  const float* P2  = (const float*)d_in[4];   // (64,144,512)
  const float* bw0 = (const float*)d_in[5];
  const float* bb0 = (const float*)d_in[6];
  const float* bw1 = (const float*)d_in[7];
  const float* bb1 = (const float*)d_in[8];
  const float* nz0 = (const float*)d_in[9];
  const float* nz1 = (const float*)d_in[10];
  float* out = (float*)d_out;

  float* ws = (float*)d_ws;
  size_t off = 0;
  float* X0   = ws + off; off += (size_t)64*256*64;    // packed input
  float* Xbuf = ws + off; off += (size_t)64*256*512;   // normalized mean halves
  float* Yb   = ws + off; off += (size_t)64*256*1024;  // GEMM out (reused L0/L1)
  float* Y2   = ws + off; off += (size_t)64*256*72;    // layer2 out
  float* ps0  = ws + off; off += 65536;
  float* pq0  = ws + off; off += 65536;
  float* ps1  = ws + off; off += 65536;
  float* pq1  = ws + off; off += 65536;
  float* sc0  = ws + off; off += 1024;
  float* sh0  = ws + off; off += 1024;
  float* sc1  = ws + off; off += 1024;
  float* sh1  = ws + off; off += 1024;
  float* Ip0  = ws + off; off += 16384;
  float* Ip1  = ws + off; off += 16384;

  pack_kernel<<<4096, 256, 0, stream>>>(Hre, Him, X0);

  // Layer 0: (1024 x 64) x (64 x 256) per n, relu
  gemm_kernel<<<dim3(4, 8, 64), 256, 0, stream>>>(P0, X0, Yb, 64, 1024, 1024, 1);
  bn_stats_kernel<<<64, 256, 0, stream>>>(Yb, ps0, pq0);
  bn_finalize_kernel<<<4, 256, 0, stream>>>(ps0, pq0, bw0, bb0, sc0, sh0);
  noise_kernel<<<dim3(64, 256), 256, 0, stream>>>(Yb, sc0, sh0, nz0, Xbuf, Ip0);

  // Layer 1: (1024 x 512) x (512 x 256) per n, relu
  gemm_kernel<<<dim3(4, 8, 64), 256, 0, stream>>>(P1, Xbuf, Yb, 512, 1024, 1024, 1);
  bn_stats_kernel<<<64, 256, 0, stream>>>(Yb, ps1, pq1);
  bn_finalize_kernel<<<4, 256, 0, stream>>>(ps1, pq1, bw1, bb1, sc1, sh1);
  noise_kernel<<<dim3(64, 256), 256, 0, stream>>>(Yb, sc1, sh1, nz1, Xbuf, Ip1);

  // Layer 2: rows 0..127 all valid (M_P=144), only first 72 outputs stored
  gemm_kernel<<<dim3(4, 1, 64), 256, 0, stream>>>(P2, Xbuf, Y2, 512, 144, 72, 0);

  final_kernel<<<256, 64, 0, stream>>>(Y2, out);
  iout_kernel<<<1, 256, 0, stream>>>(Ip0, Ip1, out + 131072);
}